// Consistent_loss_left_25288767439318
// MI455X (gfx1250) — compile-verified
//
#include <hip/hip_runtime.h>
#include <hip/hip_bf16.h>

// ---------------------------------------------------------------------------
// Consistent_loss_left — MI455X (gfx1250, wave32)
//
// reference:  scatter-max of (110-i)/50 (i<110, left>=0.0235) into row bin
//             r = round(128 - 60*left) per (b, column j), then
//             mean( |left2up - up| where diff < 0.2 )
//
// Memory-bound: reads 64 MiB of `up` + ~16 MiB of `left[:, :, :110]`,
// skips `right` entirely. ~80 MiB @ 23.3 TB/s ≈ 3.5 us floor.
// Fused single pass: LDS bins (61 rows x 32 cols, padded) + streaming compare.
// ---------------------------------------------------------------------------

#define B_  64
#define H_  512
#define W_  512
#define ICUT 110          // I_CUTOFF
#define RMIN 68           // round(128 - 60*l), l in [0,1]  ->  r in [68,128]
#define NR   61           // 128 - 68 + 1 bins
#define JT   32           // columns per block tile (128B coalesced up-reads)
#define PAD  33           // padded LDS row stride (bank-spread for scatter)
#define NT_  512          // threads per block (16 wave32)
#define NTILES (W_ / JT)                  // 16
#define NBLOCKS (B_ * NTILES)             // 1024

__global__ __launch_bounds__(NT_) void loss_stage1(
    const float* __restrict__ up, const float* __restrict__ left,
    float* __restrict__ partials)
{
  __shared__ int   bins[NR * PAD];        // 61*33*4 = 8052 B
  __shared__ float red[NT_ / 32];

  const int tid = threadIdx.x;
  const int blk = blockIdx.x;
  const int b   = blk / NTILES;
  const int j0  = (blk % NTILES) * JT;

  // zero bins (scatter-max identity)
  for (int k = tid; k < NR * PAD; k += NT_) bins[k] = 0;
  __syncthreads();

  // ---- phase 1: per-column scatter-max into LDS bins (ds_max_i32) --------
  {
    const size_t base = (size_t)b * (H_ * W_) + (size_t)j0 * W_;
    const int i = tid & 127;              // lane-contiguous -> coalesced left
    const int j = tid >> 7;               // 4 columns per pass
    if (i < ICUT) {
      const float val   = (float)(ICUT - i) / 50.0f;   // > 0, so int-ordered
      const int   vbits = __float_as_int(val);
      for (int jj = j; jj < JT; jj += NT_ >> 7) {
        float l = __builtin_nontemporal_load(&left[base + (size_t)jj * W_ + i]);
        if (l >= 0.0235f) {
          // match jnp: unfused mul/sub, round-half-even, clip
          int r = __float2int_rn(__fsub_rn(128.0f, __fmul_rn(l, 60.0f)));
          r = min(max(r, 0), H_ - 1);
          if ((unsigned)(r - RMIN) < (unsigned)NR)
            atomicMax(&bins[(r - RMIN) * PAD + jj], vbits);
        }
      }
    }
  }
  __syncthreads();

  // ---- phase 2: streaming masked-|diff| accumulation ---------------------
  float acc = 0.0f;
  {
    const int jl = tid & (JT - 1);        // lane -> column, 128B coalesced
    const int r0 = tid >> 5;              // 0..15
    const size_t base = (size_t)b * (H_ * W_) + (size_t)j0 + jl;
    for (int r = r0; r < H_; r += NT_ / JT) {
      float u = __builtin_nontemporal_load(&up[base + (size_t)r * W_]);
      float v = 0.0f;
      if ((unsigned)(r - RMIN) < (unsigned)NR)
        v = __int_as_float(bins[(r - RMIN) * PAD + jl]);
      float d = fabsf(v - u);
      if (d < 0.2f) acc += d;
    }
  }

  // ---- deterministic block reduction (wave32 shuffle tree + LDS) ---------
  for (int o = 16; o > 0; o >>= 1) acc += __shfl_down(acc, o, 32);
  if ((tid & 31) == 0) red[tid >> 5] = acc;
  __syncthreads();
  if (tid < 32) {
    float v = (tid < NT_ / 32) ? red[tid] : 0.0f;
    for (int o = 16; o > 0; o >>= 1) v += __shfl_down(v, o, 32);
    if (tid == 0) partials[blk] = v;
  }
}

// fixed-order final reduce in double, scale by exact 2^-24
__global__ __launch_bounds__(256) void loss_stage2(
    const float* __restrict__ partials, float* __restrict__ out)
{
  __shared__ double red[8];
  double acc = 0.0;
  for (int k = threadIdx.x; k < NBLOCKS; k += 256)
    acc += (double)partials[k];
  for (int o = 16; o > 0; o >>= 1) acc += __shfl_down(acc, o, 32);
  if ((threadIdx.x & 31) == 0) red[threadIdx.x >> 5] = acc;
  __syncthreads();
  if (threadIdx.x == 0) {
    double s = 0.0;
    for (int w = 0; w < 8; ++w) s += red[w];
    out[0] = (float)(s * (1.0 / (double)((size_t)B_ * H_ * W_)));
  }
}

extern "C" void kernel_launch(void* const* d_in, const int* in_sizes, int n_in,
                              void* d_out, int out_size, void* d_ws, size_t ws_size,
                              hipStream_t stream) {
  (void)in_sizes; (void)n_in; (void)out_size; (void)ws_size;
  const float* up   = (const float*)d_in[0];
  const float* left = (const float*)d_in[1];
  // d_in[2] (right) is unused by the reference -> never read (saves 64 MiB).
  float* partials = (float*)d_ws;         // NBLOCKS floats, fully rewritten
  float* out      = (float*)d_out;

  loss_stage1<<<NBLOCKS, NT_, 0, stream>>>(up, left, partials);
  loss_stage2<<<1, 256, 0, stream>>>(partials, out);
}